// GlobalAttention_27393301414156
// MI455X (gfx1250) — compile-verified
//
#include <hip/hip_runtime.h>
#include <hip/hip_bf16.h>

typedef __attribute__((ext_vector_type(2))) float v2f;
typedef __attribute__((ext_vector_type(8))) float v8f;

#define B_ 16
#define N_ 4096
#define C_ 1024
#define NCHUNK 16   // chunks per batch
#define CHUNK 256   // rows per block
#define XSTR 1032   // xs row stride (pad 8 floats -> conflict-free half-wave B reads)

// ---------------- precompute: LN of class_feature row ----------------
__global__ __launch_bounds__(256) void ga_lncf(const float* __restrict__ cf,
                                               const float* __restrict__ qg,
                                               const float* __restrict__ qb,
                                               float* __restrict__ cfn) {
  __shared__ float s0[256], s1[256];
  const int tid = threadIdx.x;
  float4 x = *(const float4*)(cf + tid * 4);
  s0[tid] = x.x + x.y + x.z + x.w;
  s1[tid] = x.x * x.x + x.y * x.y + x.z * x.z + x.w * x.w;
  __syncthreads();
  for (int s = 128; s > 0; s >>= 1) {
    if (tid < s) { s0[tid] += s0[tid + s]; s1[tid] += s1[tid + s]; }
    __syncthreads();
  }
  const float mu = s0[0] * (1.f / 1024.f);
  const float var = s1[0] * (1.f / 1024.f) - mu * mu;
  const float rs = rsqrtf(var + 1e-5f);
  float4 g = *(const float4*)(qg + tid * 4);
  float4 b4 = *(const float4*)(qb + tid * 4);
  float4 o;
  o.x = (x.x - mu) * rs * g.x + b4.x;
  o.y = (x.y - mu) * rs * g.y + b4.y;
  o.z = (x.z - mu) * rs * g.z + b4.z;
  o.w = (x.w - mu) * rs * g.w + b4.w;
  *(float4*)(cfn + tid * 4) = o;
}

// ---------------- precompute: qs = (cfn @ Wq^T) * C^-0.5 ----------------
__global__ __launch_bounds__(256) void ga_q(const float* __restrict__ cfn,
                                            const float* __restrict__ Wq,
                                            float* __restrict__ qs) {
  __shared__ float cs[1024];
  const int tid = threadIdx.x;
  for (int i = tid; i < 1024; i += 256) cs[i] = cfn[i];
  __syncthreads();
  const int wv = tid >> 5, ln = tid & 31;
  for (int jj = 0; jj < 32; ++jj) {
    const int j = blockIdx.x * 256 + wv * 32 + jj;
    const float* wrow = Wq + (size_t)j * 1024;
    float acc = 0.f;
    for (int c0 = 0; c0 < 1024; c0 += 32) acc += cs[c0 + ln] * wrow[c0 + ln];
#pragma unroll
    for (int m = 16; m > 0; m >>= 1) acc += __shfl_xor(acc, m, 32);
    if (ln == 0) qs[j] = acc * 0.03125f;  // scale = 1024^-0.5
  }
}

// ------ precompute: wqeffT[c][h] = sum_d Wkv_k[h*128+d, c] * qs[h*128+d], padded to 16 heads ------
__global__ __launch_bounds__(256) void ga_wqeff(const float* __restrict__ Wkv,
                                                const float* __restrict__ qs,
                                                float* __restrict__ wqe) {
  __shared__ float q_s[1024];
  const int tid = threadIdx.x;
  for (int i = tid; i < 1024; i += 256) q_s[i] = qs[i];
  __syncthreads();
  const int c = blockIdx.x * 256 + tid;
  for (int h = 0; h < 8; ++h) {
    float acc = 0.f;
    for (int d = 0; d < 128; ++d)
      acc += Wkv[(size_t)(h * 128 + d) * 1024 + c] * q_s[h * 128 + d];
    wqe[c * 16 + h] = acc;
  }
  for (int h = 8; h < 16; ++h) wqe[c * 16 + h] = 0.f;
}

// ---------------- main fused LN + flash-attention partial kernel ----------------
__global__ __launch_bounds__(256) void ga_main(const float* __restrict__ feats,
                                               const float* __restrict__ kvg,
                                               const float* __restrict__ kvb,
                                               const float* __restrict__ wqe,
                                               float* __restrict__ mp,
                                               float* __restrict__ lp,
                                               float* __restrict__ yp) {
  __shared__ __align__(16) float xs[16][XSTR];     // normalized feature tile
  __shared__ __align__(16) float wq[1024 * 17];    // wqeffT, padded stride
  __shared__ __align__(32) float sred[2048];       // per-wave s partials / LN partials
  __shared__ float sS[256];                        // s tile [row][head]
  __shared__ float sP[16 * 17];                    // P tile [row][head], padded
  __shared__ float sF[16];                         // per-head rescale factors
  __shared__ float smu[16], srs[16];

  const int tid = threadIdx.x;
  const int lane = tid & 31;
  const int wave = tid >> 5;
  const int l16 = lane & 15;
  const int lhi = lane >> 4;
  const int r = tid >> 4;    // row within subtile (load phase)
  const int t16 = tid & 15;  // column group within row (load phase)

  const int blk = blockIdx.x;
  const int bb = blk >> 4;
  const int chunk = blk & 15;

  for (int i = tid; i < 16384; i += 256) wq[(i >> 4) * 17 + (i & 15)] = wqe[i];

  const v8f vzero = {0.f, 0.f, 0.f, 0.f, 0.f, 0.f, 0.f, 0.f};
  v8f yacc[8];
#pragma unroll
  for (int t = 0; t < 8; ++t) yacc[t] = vzero;

  float m_run = -3.4e38f;
  float l_run = 0.f;

  __syncthreads();

#pragma unroll 1
  for (int st = 0; st < 16; ++st) {
    // ---- load 16 rows, compute LayerNorm into xs ----
    const size_t rowbase =
        ((size_t)bb * N_ + (size_t)chunk * CHUNK + st * 16 + r) * C_;
    float sum = 0.f, sq = 0.f;
#pragma unroll
    for (int i = 0; i < 16; ++i) {
      const int col = i * 64 + t16 * 4;
      float4 x = *(const float4*)(feats + rowbase + col);
      sum += x.x + x.y + x.z + x.w;
      sq += x.x * x.x + x.y * x.y + x.z * x.z + x.w * x.w;
      *(float4*)&xs[r][col] = x;
    }
    if (st < 15)  // prefetch next subtile (global_prefetch_b8)
      __builtin_prefetch(feats + rowbase + 16 * C_ + t16 * 64, 0, 3);
    sred[tid] = sum;
    sred[256 + tid] = sq;
    __syncthreads();
    if (tid < 16) {
      float s0 = 0.f, s1 = 0.f;
#pragma unroll
      for (int i = 0; i < 16; ++i) {
        s0 += sred[tid * 16 + i];
        s1 += sred[256 + tid * 16 + i];
      }
      const float mu = s0 * (1.f / 1024.f);
      const float var = s1 * (1.f / 1024.f) - mu * mu;
      smu[tid] = mu;
      srs[tid] = rsqrtf(var + 1e-5f);
    }
    __syncthreads();
    {
      const float mu = smu[r], rs = srs[r];
#pragma unroll
      for (int i = 0; i < 16; ++i) {
        const int col = i * 64 + t16 * 4;
        float4 x = *(float4*)&xs[r][col];
        float4 g = *(const float4*)(kvg + col);
        float4 b4 = *(const float4*)(kvb + col);
        x.x = (x.x - mu) * rs * g.x + b4.x;
        x.y = (x.y - mu) * rs * g.y + b4.y;
        x.z = (x.z - mu) * rs * g.z + b4.z;
        x.w = (x.w - mu) * rs * g.w + b4.w;
        *(float4*)&xs[r][col] = x;
      }
    }
    __syncthreads();

    // ---- s = xn @ wqeffT via fp32 WMMA; each wave owns a K-slice of 128 ----
    v8f sacc = vzero;
    const int kb = wave * 128 + 2 * lhi;
#pragma unroll
    for (int kk = 0; kk < 32; ++kk) {
      const int k0 = kb + kk * 4;
      v2f a = *(const v2f*)&xs[l16][k0];  // A[m=row][k]
      v2f bv;                             // B[k][n=head]
      bv.x = wq[k0 * 17 + l16];
      bv.y = wq[(k0 + 1) * 17 + l16];
      sacc = __builtin_amdgcn_wmma_f32_16x16x4_f32(false, a, false, bv,
                                                   (short)0, sacc, false, false);
    }
    *(v8f*)&sred[wave * 256 + lane * 8] = sacc;
    __syncthreads();
    {  // cross-wave reduction of the 16x16 s tile
      const int rr = tid >> 4, hh = tid & 15;
      const int idx = (hh + ((rr >> 3) << 4)) * 8 + (rr & 7);
      float v = 0.f;
#pragma unroll
      for (int w = 0; w < 8; ++w) v += sred[w * 256 + idx];
      sS[rr * 16 + hh] = v;
    }
    __syncthreads();

    // ---- online softmax update: one thread per (padded) head ----
    if (tid < 16) {
      float rmax = -3.4e38f;
#pragma unroll
      for (int q = 0; q < 16; ++q) rmax = fmaxf(rmax, sS[q * 16 + tid]);
      const float mnew = fmaxf(m_run, rmax);
      const float f = __expf(m_run - mnew);
      float ls = 0.f;
#pragma unroll
      for (int q = 0; q < 16; ++q) {
        const float p = __expf(sS[q * 16 + tid] - mnew);
        sP[q * 17 + tid] = p;
        ls += p;
      }
      l_run = l_run * f + ls;
      m_run = mnew;
      sF[tid] = f;
    }
    __syncthreads();

    // ---- rescale Y, then Y += P^T @ xn via WMMA; wave owns 128 columns ----
    float fv[8];
#pragma unroll
    for (int v = 0; v < 8; ++v) fv[v] = sF[v + 8 * lhi];
#pragma unroll
    for (int t = 0; t < 8; ++t) {
#pragma unroll
      for (int v = 0; v < 8; ++v) yacc[t][v] *= fv[v];
    }
#pragma unroll
    for (int k0 = 0; k0 < 16; k0 += 4) {
      const int kr = k0 + 2 * lhi;
      v2f a;  // A[m=head][k=row] = P[row][head]
      a.x = sP[kr * 17 + l16];
      a.y = sP[(kr + 1) * 17 + l16];
#pragma unroll
      for (int t = 0; t < 8; ++t) {
        const int c0 = wave * 128 + t * 16;
        v2f bv;  // B[k=row][n=col]
        bv.x = xs[kr][c0 + l16];
        bv.y = xs[kr + 1][c0 + l16];
        yacc[t] = __builtin_amdgcn_wmma_f32_16x16x4_f32(
            false, a, false, bv, (short)0, yacc[t], false, false);
      }
    }
    __syncthreads();
  }

  // ---- write per-chunk partials (heads 0..7 live in lanes 0..15) ----
  if (tid < 8) {
    mp[blk * 8 + tid] = m_run;
    lp[blk * 8 + tid] = l_run;
  }
  if (lhi == 0) {
#pragma unroll
    for (int t = 0; t < 8; ++t) {
      const int c0 = wave * 128 + t * 16;
#pragma unroll
      for (int v = 0; v < 8; ++v)
        yp[(size_t)(blk * 8 + v) * 1024 + c0 + l16] = yacc[t][v];
    }
  }
}

// ---------------- merge partials + apply Wv + output projection ----------------
__global__ __launch_bounds__(1024) void ga_merge(const float* __restrict__ mp,
                                                 const float* __restrict__ lp,
                                                 const float* __restrict__ yp,
                                                 const float* __restrict__ Wkv,
                                                 const float* __restrict__ projW,
                                                 const float* __restrict__ projb,
                                                 const float* __restrict__ cf,
                                                 float* __restrict__ out) {
  __shared__ float yn[8 * 1024];
  __shared__ float ag[1024];
  __shared__ float ef[16 * 8];
  __shared__ float Lh[8];
  const int tid = threadIdx.x;
  const int b = blockIdx.x;
  if (tid < 8) {
    float M = -3.4e38f;
    for (int k = 0; k < 16; ++k) M = fmaxf(M, mp[(b * 16 + k) * 8 + tid]);
    float L = 0.f;
    for (int k = 0; k < 16; ++k) {
      const float e = __expf(mp[(b * 16 + k) * 8 + tid] - M);
      ef[k * 8 + tid] = e;
      L += lp[(b * 16 + k) * 8 + tid] * e;
    }
    Lh[tid] = L;
  }
  __syncthreads();
  for (int e = tid; e < 8192; e += 1024) {
    const int h = e >> 10, c = e & 1023;
    float acc = 0.f;
    for (int k = 0; k < 16; ++k)
      acc += yp[(size_t)((b * 16 + k) * 8 + h) * 1024 + c] * ef[k * 8 + h];
    yn[e] = acc / Lh[h];
  }
  __syncthreads();
  const int wv = tid >> 5, ln = tid & 31;
  for (int jj = 0; jj < 32; ++jj) {  // agg[j] = yn[h] . Wv[j]
    const int j = wv * 32 + jj;
    const int h = j >> 7;
    const float* wrow = Wkv + (size_t)(1024 + j) * 1024;
    float acc = 0.f;
    for (int c0 = 0; c0 < 1024; c0 += 32) acc += yn[h * 1024 + c0 + ln] * wrow[c0 + ln];
#pragma unroll
    for (int m = 16; m > 0; m >>= 1) acc += __shfl_xor(acc, m, 32);
    if (ln == 0) ag[j] = acc;
  }
  __syncthreads();
  for (int jj = 0; jj < 32; ++jj) {  // out = cf + agg @ projW^T + projb
    const int j = wv * 32 + jj;
    const float* prow = projW + (size_t)j * 1024;
    float acc = 0.f;
    for (int c0 = 0; c0 < 1024; c0 += 32) acc += ag[c0 + ln] * prow[c0 + ln];
#pragma unroll
    for (int m = 16; m > 0; m >>= 1) acc += __shfl_xor(acc, m, 32);
    if (ln == 0) out[(size_t)b * 1024 + j] = cf[j] + acc + projb[j];
  }
}

extern "C" void kernel_launch(void* const* d_in, const int* in_sizes, int n_in,
                              void* d_out, int out_size, void* d_ws, size_t ws_size,
                              hipStream_t stream) {
  const float* class_feature = (const float*)d_in[0];
  const float* features      = (const float*)d_in[1];
  const float* q_gamma       = (const float*)d_in[2];
  const float* q_beta        = (const float*)d_in[3];
  const float* Wq            = (const float*)d_in[4];
  const float* kv_gamma      = (const float*)d_in[5];
  const float* kv_beta       = (const float*)d_in[6];
  const float* Wkv           = (const float*)d_in[7];
  const float* proj_W        = (const float*)d_in[8];
  const float* proj_b        = (const float*)d_in[9];
  float* out = (float*)d_out;

  float* ws   = (float*)d_ws;
  float* cfn  = ws;               // 1024
  float* qs   = ws + 1024;        // 1024
  float* wqe  = ws + 2048;        // 1024*16
  float* mp   = ws + 18432;       // B*16*8
  float* lp   = ws + 20480;       // B*16*8
  float* yp   = ws + 22528;       // B*16*8*1024 = 2097152

  ga_lncf<<<1, 256, 0, stream>>>(class_feature, q_gamma, q_beta, cfn);
  ga_q<<<4, 256, 0, stream>>>(cfn, Wq, qs);
  ga_wqeff<<<4, 256, 0, stream>>>(Wkv, qs, wqe);
  ga_main<<<B_ * NCHUNK, 256, 0, stream>>>(features, kv_gamma, kv_beta, wqe, mp, lp, yp);
  ga_merge<<<B_, 1024, 0, stream>>>(mp, lp, yp, Wkv, proj_W, proj_b, class_feature, out);
}